// MetaOptNet_78151224918357
// MI455X (gfx1250) — compile-verified
//
#include <hip/hip_runtime.h>
#include <math.h>

typedef __attribute__((ext_vector_type(2))) float v2f;
typedef __attribute__((ext_vector_type(8))) float v8f;

#define D_DIM   4096
#define N_WAY   5
#define N_SUP   5
#define N_QRY   3000
#define NS      25          // support count
#define NN      125         // QP vars (n)
#define NP      25          // equality constraints (p)
#define NT      150         // KKT size
#define NC      151         // KKT cols incl. RHS
#define NQ_TOT  15000       // total query rows
#define SIGMA_F 0.1f
#define CREG_F  0.1f

// row of x holding support sample j (j = class*5 + k)
__device__ __forceinline__ int support_xrow(int j) {
    return (j / N_SUP) * (N_SUP + N_QRY) + (j % N_SUP);
}

// ---------------------------------------------------------------------------
// Kernel 1: K = support @ support^T   (25x25), one block per (i,j)
// ---------------------------------------------------------------------------
__global__ void k_gram(const float* __restrict__ x, float* __restrict__ Kmat) {
    int b = blockIdx.x;               // 0..624
    int i = b / NS, j = b % NS;
    const float* pi = x + (long)support_xrow(i) * D_DIM;
    const float* pj = x + (long)support_xrow(j) * D_DIM;
    float sum = 0.f;
    for (int k = threadIdx.x; k < D_DIM; k += 256) sum += pi[k] * pj[k];
    __shared__ float red[256];
    red[threadIdx.x] = sum;
    __syncthreads();
    for (int st = 128; st; st >>= 1) {
        if (threadIdx.x < st) red[threadIdx.x] += red[threadIdx.x + st];
        __syncthreads();
    }
    if (threadIdx.x == 0) Kmat[b] = red[0];
}

// ---------------------------------------------------------------------------
// Kernel 2: interior-point QP, single workgroup, 15 iterations.
// KKT (150x151 augmented) lives in global scratch; vectors in LDS.
// ---------------------------------------------------------------------------
__global__ void __launch_bounds__(256) k_qp(const float* __restrict__ Kmat,
                                            float* __restrict__ KKT,
                                            float* __restrict__ zout) {
    __shared__ float K[NS * NS];
    __shared__ float z[NN], s[NN], lam[NN], nu[NP];
    __shared__ float rd[NN], rp[NN], rq[NP], rc[NN], wv[NN], rz[NN];
    __shared__ float fac[NT], sol[NT], dsv[NN], dlv[NN];
    __shared__ float red[256];
    __shared__ float mu_s, alpha_s;

    const int t = threadIdx.x;
    for (int i = t; i < NS * NS; i += 256) K[i] = Kmat[i];
    for (int i = t; i < NN; i += 256) { z[i] = 0.f; s[i] = 1.f; lam[i] = 1.f; }
    for (int i = t; i < NP; i += 256) nu[i] = 0.f;
    __syncthreads();

    for (int iter = 0; iter < 15; ++iter) {
        // ---- residuals + mu partials -------------------------------------
        float part = 0.f;
        for (int i = t; i < NN; i += 256) {
            int a = i / N_WAY, c = i % N_WAY;           // support idx, class slot
            float gz = z[i];                            // + I z
            for (int b = 0; b < NS; ++b) gz += K[a * NS + b] * z[N_WAY * b + c];
            float y1 = (c == a / N_SUP) ? 1.f : 0.f;
            rd[i] = gz - y1 + lam[i] + nu[a];           // e = -y1, C^T lam = lam
            rp[i] = z[i] + s[i] - CREG_F * y1;          // h = 0.1*y1
            part += s[i] * lam[i];
        }
        for (int a = t; a < NP; a += 256) {
            float sum = 0.f;
            for (int c = 0; c < N_WAY; ++c) sum += z[N_WAY * a + c];
            rq[a] = sum;                                // A z  (b = 0)
        }
        red[t] = part;
        __syncthreads();
        if (t == 0) {
            float m = 0.f;
            for (int k = 0; k < 256; ++k) m += red[k];
            mu_s = m / (float)NN;
        }
        __syncthreads();

        // ---- centering terms ---------------------------------------------
        for (int i = t; i < NN; i += 256) {
            float rcv = s[i] * lam[i] - SIGMA_F * mu_s;
            rc[i] = rcv;
            wv[i] = lam[i] / s[i];
            rz[i] = -(rd[i] + (-rcv + lam[i] * rp[i]) / s[i]);
        }
        __syncthreads();

        // ---- build augmented KKT  [[M, A^T],[A,0]] | [rhs_z, -r_peq] -----
        for (int idx = t; idx < NT * NC; idx += 256) {
            int r = idx / NC, c = idx % NC;
            float v;
            if (r < NN) {
                if (c < NN) {
                    int a = r / N_WAY, rr = r % N_WAY, b = c / N_WAY, cc = c % N_WAY;
                    v = (rr == cc) ? K[a * NS + b] : 0.f;     // kron(K, I5)
                    if (r == c) v += 1.f + wv[r];             // + I + diag(w)
                } else if (c < NT) {
                    v = ((c - NN) == r / N_WAY) ? 1.f : 0.f;  // A^T
                } else {
                    v = rz[r];
                }
            } else {
                int a = r - NN;
                if (c < NN)       v = (c / N_WAY == a) ? 1.f : 0.f;  // A
                else if (c < NT)  v = 0.f;
                else              v = -rq[a];
            }
            KKT[idx] = v;
        }
        __syncthreads();

        // ---- Gaussian elimination (no pivoting; pivots provably nonzero) --
        for (int k = 0; k < NT - 1; ++k) {
            float piv = KKT[k * NC + k];
            for (int r = k + 1 + t; r < NT; r += 256) fac[r] = KKT[r * NC + k] / piv;
            __syncthreads();
            int nr = NT - 1 - k, ncol = NC - 1 - k;
            for (int idx = t; idx < nr * ncol; idx += 256) {
                int r = k + 1 + idx / ncol;
                int c = k + 1 + idx % ncol;
                KKT[r * NC + c] -= fac[r] * KKT[k * NC + c];
            }
            __syncthreads();
        }

        // ---- back substitution on wave 0 ----------------------------------
        if (t < 32) {
            volatile float* vsol = sol;
            for (int r = NT - 1; r >= 0; --r) {
                float sum = 0.f;
                for (int c = r + 1 + t; c < NT; c += 32) sum += KKT[r * NC + c] * vsol[c];
                for (int off = 16; off; off >>= 1) sum += __shfl_down(sum, off, 32);
                if (t == 0) vsol[r] = (KKT[r * NC + NT] - sum) / KKT[r * NC + r];
            }
        }
        __syncthreads();

        // ---- step directions & fraction-to-boundary -----------------------
        for (int i = t; i < NN; i += 256) {
            float dz = sol[i];
            float ds = -rp[i] - dz;
            float dl = (-rc[i] - lam[i] * ds) / s[i];
            dsv[i] = ds;
            dlv[i] = dl;
        }
        __syncthreads();
        if (t == 0) {
            float as = INFINITY, al = INFINITY;
            for (int i = 0; i < NN; ++i) {
                if (dsv[i] < 0.f) as = fminf(as, -s[i] / dsv[i]);
                if (dlv[i] < 0.f) al = fminf(al, -lam[i] / dlv[i]);
            }
            alpha_s = fminf(1.f, 0.99f * fminf(as, al));
        }
        __syncthreads();
        float alpha = alpha_s;
        for (int i = t; i < NN; i += 256) {
            z[i]   += alpha * sol[i];
            s[i]   += alpha * dsv[i];
            lam[i] += alpha * dlv[i];
        }
        for (int i = t; i < NP; i += 256) nu[i] += alpha * sol[NN + i];
        __syncthreads();
    }

    for (int i = t; i < NN; i += 256) zout[i] = z[i];
}

// ---------------------------------------------------------------------------
// Kernel 3: W[d, c] = sum_j support[j,d] * qp[j,c].
// Stored pair-interleaved for b64 B-fetches in the WMMA kernel:
//   Wb[(d/2)*32 + c*2 + (d&1)] = W[d][c]   (c in [0,16), zero-padded c>=5)
// ---------------------------------------------------------------------------
__global__ void k_weights(const float* __restrict__ x, const float* __restrict__ z,
                          float* __restrict__ Wb) {
    int d = blockIdx.x * 256 + threadIdx.x;
    if (d >= D_DIM) return;
    float acc[N_WAY] = {0.f, 0.f, 0.f, 0.f, 0.f};
    for (int j = 0; j < NS; ++j) {
        float sv = x[(long)support_xrow(j) * D_DIM + d];
        for (int c = 0; c < N_WAY; ++c) acc[c] += sv * z[N_WAY * j + c];
    }
    float* base = Wb + (d >> 1) * 32 + (d & 1);
    for (int c = 0; c < 16; ++c) base[c * 2] = (c < N_WAY) ? acc[c] : 0.f;
}

// ---------------------------------------------------------------------------
// Kernel 4: logits = query @ W  via V_WMMA_F32_16X16X4_F32.
// One wave per 16-row M-tile; K stepped by 4; N padded 5 -> 16.
// A layout (16x4 f32): lane l holds M=l%16, VGPR v holds K = 2*(l/16)+v.
// B layout (4x16 f32): lane l holds N=l%16, VGPR v holds K = 2*(l/16)+v.
// C/D layout: VGPR r, lane l -> M = r + 8*(l/16), N = l%16.
// Invalid (padded) M rows are clamped to row NQ_TOT-1: their outputs are never stored
// and row M of D depends only on row M of A, so valid rows are unaffected.
// ---------------------------------------------------------------------------
__global__ void __launch_bounds__(256) k_logits(const float* __restrict__ x,
                                                const float* __restrict__ Wb,
                                                float* __restrict__ out) {
    const int wave = threadIdx.x >> 5;
    const int lane = threadIdx.x & 31;
    const int tile = blockIdx.x * 8 + wave;
    const int mbase = tile * 16;
    if (mbase >= NQ_TOT) return;                 // uniform per wave

    const int mrow  = lane & 15;                 // M within tile (also N for B)
    const int khalf = lane >> 4;                 // 0/1 -> K offset 2*khalf
    const int q     = min(mbase + mrow, NQ_TOT - 1);   // clamp padded rows
    const int cls = q / N_QRY, within = q % N_QRY;
    const long xrow = (long)cls * (N_SUP + N_QRY) + N_SUP + within;
    const float* arow = x + xrow * D_DIM + 2 * khalf;
    // pair index k' = k0/2 + khalf; addr = (k'*16 + n)*2 = k0*16 + khalf*32 + 2n
    const float* brow = Wb + khalf * 32 + 2 * mrow;

    v8f acc = {};
#pragma unroll 4
    for (int k0 = 0; k0 < D_DIM; k0 += 4) {
        v2f a = *(const v2f*)(arow + k0);        // b64: A[M, K..K+1]
        v2f b = *(const v2f*)(brow + k0 * 16);   // b64: W[K][n], W[K+1][n]
        acc = __builtin_amdgcn_wmma_f32_16x16x4_f32(
            /*neg_a=*/false, a, /*neg_b=*/false, b,
            /*c_mod=*/(short)0, acc, /*reuse_a=*/false, /*reuse_b=*/false);
    }

    const int n = lane & 15;
    const int rbase = 8 * (lane >> 4);
    if (n < N_WAY) {
        float* obase = out + (long)(mbase + rbase) * N_WAY + n;
        if (mbase + 16 <= NQ_TOT) {              // full tile: unguarded stores
#pragma unroll
            for (int r = 0; r < 8; ++r) obase[r * N_WAY] = acc[r];
        } else {
#pragma unroll
            for (int r = 0; r < 8; ++r) {
                if (mbase + rbase + r < NQ_TOT) obase[r * N_WAY] = acc[r];
            }
        }
    }
}

// ---------------------------------------------------------------------------
extern "C" void kernel_launch(void* const* d_in, const int* in_sizes, int n_in,
                              void* d_out, int out_size, void* d_ws, size_t ws_size,
                              hipStream_t stream) {
    const float* x = (const float*)d_in[0];
    float* out = (float*)d_out;
    float* ws  = (float*)d_ws;

    float* Kmat = ws;              // 625 floats
    float* zbuf = ws + 704;        // 125 floats
    float* KKT  = ws + 1024;       // 150*151 = 22650 floats
    float* Wb   = ws + 24064;      // 4096*16 = 65536 floats (pair-interleaved)

    k_gram   <<<NS * NS, 256, 0, stream>>>(x, Kmat);
    k_qp     <<<1,        256, 0, stream>>>(Kmat, KKT, zbuf);
    k_weights<<<D_DIM / 256, 256, 0, stream>>>(x, zbuf, Wb);
    int tiles  = (NQ_TOT + 15) / 16;           // 938
    int blocks = (tiles + 7) / 8;              // 118
    k_logits <<<blocks,   256, 0, stream>>>(x, Wb, out);
}